// StructuralAttentionLayer_11751030522841
// MI455X (gfx1250) — compile-verified
//
#include <hip/hip_runtime.h>
#include <math.h>

typedef float v2f __attribute__((ext_vector_type(2)));
typedef float v8f __attribute__((ext_vector_type(8)));

__device__ __forceinline__ v8f wmma4(v2f a, v2f b, v8f c) {
  // D = A(16x4 f32) * B(4x16 f32) + C, fp32 accumulate
  return __builtin_amdgcn_wmma_f32_16x16x4_f32(false, a, false, b, (short)0, c, false, false);
}

__device__ __forceinline__ v8f vzero8() { v8f z = {0.f,0.f,0.f,0.f,0.f,0.f,0.f,0.f}; return z; }

// ---------------------------------------------------------------------------
// Pad: build zero-padded W_rel_attn (64 x 132 x 144) and the bias/zero columns
// of rel_d1 / rel_h1 (2048 x 132, cols 128..131 = {1,0,0,0}).
// ---------------------------------------------------------------------------
__global__ void pad_kernel(const float* __restrict__ Wr, float* __restrict__ Wp,
                           float* __restrict__ Dp, float* __restrict__ Hp) {
  int idx = blockIdx.x * 256 + threadIdx.x;
  if (idx < 64 * 132 * 144) {
    int r = idx / (132 * 144);
    int rem = idx - r * (132 * 144);
    int i = rem / 144, j = rem - i * 144;
    Wp[idx] = (i < 129 && j < 129) ? Wr[((size_t)r * 129 + i) * 129 + j] : 0.f;
  }
  if (idx < 2048 * 4) {
    int row = idx >> 2, c = 128 + (idx & 3);
    float v = (c == 128) ? 1.f : 0.f;
    Dp[(size_t)row * 132 + c] = v;
    Hp[(size_t)row * 132 + c] = v;
  }
}

// ---------------------------------------------------------------------------
// MLP GEMM: Y[m,n] = act(sum_k X[m,k] * W[n,k] + b[n]), act = leaky_relu(0.1)
// One wave per 16x16 tile, K in steps of 4 via V_WMMA_F32_16X16X4_F32.
// ---------------------------------------------------------------------------
__global__ void mlp_gemm(const float* __restrict__ X, const float* __restrict__ W,
                         const float* __restrict__ bias, float* __restrict__ Y,
                         int K, int ldY, int leaky) {
  int m0 = blockIdx.x * 16, n0 = blockIdx.y * 16;
  int lane = threadIdx.x & 31;
  int half = lane >> 4, l15 = lane & 15;
  const float* xrow = X + (size_t)(m0 + l15) * K;
  const float* wrow = W + (size_t)(n0 + l15) * K;
  v8f acc = vzero8();
  for (int k0 = 0; k0 < K; k0 += 4) {
    int ka = k0 + half * 2;
    v2f a = { xrow[ka], xrow[ka + 1] };
    v2f b = { wrow[ka], wrow[ka + 1] };
    acc = wmma4(a, b, acc);
  }
#pragma unroll
  for (int v = 0; v < 8; ++v) {
    int m = m0 + v + half * 8;
    int n = n0 + l15;
    float val = acc[v] + bias[n];
    if (leaky) val = (val > 0.f) ? val : 0.1f * val;
    Y[(size_t)m * ldY + n] = val;
  }
}

// ---------------------------------------------------------------------------
// U = arc_d1 @ W_arc_attn : U[m,n] = sum_{i<512} arc_d[m,i]*Wk[i,n] + Wk[512,n]
// Wk is (513,512), k-major.
// ---------------------------------------------------------------------------
__global__ void u_gemm(const float* __restrict__ X, const float* __restrict__ Wk,
                       float* __restrict__ U) {
  int m0 = blockIdx.x * 16, n0 = blockIdx.y * 16;
  int lane = threadIdx.x & 31;
  int half = lane >> 4, l15 = lane & 15;
  const float* xrow = X + (size_t)(m0 + l15) * 512;
  v8f acc = vzero8();
  for (int k0 = 0; k0 < 512; k0 += 4) {
    int ka = k0 + half * 2;
    v2f a = { xrow[ka], xrow[ka + 1] };
    v2f b = { Wk[(size_t)ka * 512 + n0 + l15], Wk[(size_t)(ka + 1) * 512 + n0 + l15] };
    acc = wmma4(a, b, acc);
  }
#pragma unroll
  for (int v = 0; v < 8; ++v) {
    int m = m0 + v + half * 8;
    int n = n0 + l15;
    U[(size_t)m * 512 + n] = acc[v] + Wk[(size_t)512 * 512 + n];
  }
}

// ---------------------------------------------------------------------------
// s_arc[b,x,y] = sum_i U[b,x,i]*arc_h[b,y,i]; masked columns -> -1e9
// ---------------------------------------------------------------------------
__global__ void sarc_gemm(const float* __restrict__ U, const float* __restrict__ arc_h,
                          const int* __restrict__ mask, float* __restrict__ s_arc) {
  int b = blockIdx.z;
  int m0 = blockIdx.x * 16, n0 = blockIdx.y * 16;
  int lane = threadIdx.x & 31;
  int half = lane >> 4, l15 = lane & 15;
  const float* urow = U + ((size_t)b * 512 + m0 + l15) * 512;
  const float* hrow = arc_h + ((size_t)b * 512 + n0 + l15) * 512;
  v8f acc = vzero8();
  for (int k0 = 0; k0 < 512; k0 += 4) {
    int ka = k0 + half * 2;
    v2f a = { urow[ka], urow[ka + 1] };
    v2f bb = { hrow[ka], hrow[ka + 1] };
    acc = wmma4(a, bb, acc);
  }
#pragma unroll
  for (int v = 0; v < 8; ++v) {
    int m = m0 + v + half * 8;
    int y = n0 + l15;
    float val = mask[b * 512 + y] ? acc[v] : -1e9f;
    s_arc[((size_t)b * 512 + m) * 512 + y] = val;
  }
}

// ---------------------------------------------------------------------------
// p_arc[b,x,:] = softmax_y(s_arc[b,x,:]) * mask[b,x]
// ---------------------------------------------------------------------------
__global__ void softmax_arc(const float* __restrict__ s_arc, const int* __restrict__ mask,
                            float* __restrict__ p_arc) {
  __shared__ float red[256];
  int row = blockIdx.x;
  int tid = threadIdx.x;
  const float* src = s_arc + (size_t)row * 512;
  float v0 = src[tid], v1 = src[tid + 256];
  red[tid] = fmaxf(v0, v1);
  __syncthreads();
  for (int s = 128; s > 0; s >>= 1) {
    if (tid < s) red[tid] = fmaxf(red[tid], red[tid + s]);
    __syncthreads();
  }
  float M = red[0];
  __syncthreads();
  float e0 = __expf(v0 - M), e1 = __expf(v1 - M);
  red[tid] = e0 + e1;
  __syncthreads();
  for (int s = 128; s > 0; s >>= 1) {
    if (tid < s) red[tid] += red[tid + s];
    __syncthreads();
  }
  float scale = (mask[row] ? 1.f : 0.f) / red[0];
  p_arc[(size_t)row * 512 + tid] = e0 * scale;
  p_arc[(size_t)row * 512 + tid + 256] = e1 * scale;
}

// ---------------------------------------------------------------------------
// T[row, r, j] = sum_i rel_d1p[row,i] * Wp[r,i,j]; K=132 padded, j stride 144.
// Grid: (row-tiles=128, r=64), one wave per block, 9 N-tiles of 16.
// ---------------------------------------------------------------------------
__global__ void t_gemm(const float* __restrict__ Dp, const float* __restrict__ Wp,
                       float* __restrict__ T) {
  int m0 = blockIdx.x * 16;
  int r = blockIdx.y;
  int lane = threadIdx.x & 31;
  int half = lane >> 4, l15 = lane & 15;
  const float* drow = Dp + (size_t)(m0 + l15) * 132;
  const float* wr = Wp + (size_t)r * 132 * 144;
  v8f acc[9];
#pragma unroll
  for (int n = 0; n < 9; ++n) acc[n] = vzero8();
  for (int k0 = 0; k0 < 132; k0 += 4) {
    int ka = k0 + half * 2;
    v2f a = { drow[ka], drow[ka + 1] };
#pragma unroll
    for (int n = 0; n < 9; ++n) {
      int j = n * 16 + l15;
      v2f b = { wr[(size_t)ka * 144 + j], wr[(size_t)(ka + 1) * 144 + j] };
      acc[n] = wmma4(a, b, acc[n]);
    }
  }
#pragma unroll
  for (int n = 0; n < 9; ++n)
#pragma unroll
    for (int v = 0; v < 8; ++v) {
      int row = m0 + v + half * 8;
      int j = n * 16 + l15;
      T[((size_t)row * 64 + r) * 144 + j] = acc[n][v];
    }
}

// ---------------------------------------------------------------------------
// Fused: per (b,x): s_rel[b,x,y,r] (raw scores, written once, contiguous),
// softmax over r in registers, Asum[b,x,r] = sum_y p_rel * p_arc.
// 8 waves, each owns y-tiles; T slice (64x132) staged in LDS.
// ---------------------------------------------------------------------------
#define LDST 132
__global__ __launch_bounds__(256) void srel_fused(
    const float* __restrict__ Hp, const float* __restrict__ T,
    const float* __restrict__ p_arc, float* __restrict__ s_rel,
    float* __restrict__ Asum) {
  __shared__ float t_lds[64 * LDST];
  __shared__ float pa_lds[512];
  __shared__ float as_lds[64];
  int row = blockIdx.x;          // b*512 + x
  int b = row >> 9;
  int tid = threadIdx.x;

  const float* tg = T + (size_t)row * 64 * 144;
  for (int idx = tid; idx < 64 * 132; idx += 256) {
    int r = idx / 132, j = idx - r * 132;
    t_lds[r * LDST + j] = tg[(size_t)r * 144 + j];
  }
  for (int idx = tid; idx < 512; idx += 256)
    pa_lds[idx] = p_arc[(size_t)row * 512 + idx];
  if (tid < 64) as_lds[tid] = 0.f;
  __syncthreads();

  int wave = tid >> 5;
  int lane = tid & 31;
  int half = lane >> 4, l15 = lane & 15;
  const float* hbase = Hp + (size_t)(b * 512) * 132;
  size_t srbase = (size_t)row * 512 * 64;
  float accA0 = 0.f, accA1 = 0.f, accA2 = 0.f, accA3 = 0.f;

  for (int yt = wave; yt < 32; yt += 8) {
    int y0 = yt * 16;
    v8f acc0 = vzero8(), acc1 = vzero8(), acc2 = vzero8(), acc3 = vzero8();
    const float* hrow = hbase + (size_t)(y0 + l15) * 132;
    for (int k0 = 0; k0 < 132; k0 += 4) {
      int ka = k0 + half * 2;
      v2f a = { hrow[ka], hrow[ka + 1] };
      v2f b0 = { t_lds[(0 * 16 + l15) * LDST + ka], t_lds[(0 * 16 + l15) * LDST + ka + 1] };
      v2f b1 = { t_lds[(1 * 16 + l15) * LDST + ka], t_lds[(1 * 16 + l15) * LDST + ka + 1] };
      v2f b2 = { t_lds[(2 * 16 + l15) * LDST + ka], t_lds[(2 * 16 + l15) * LDST + ka + 1] };
      v2f b3 = { t_lds[(3 * 16 + l15) * LDST + ka], t_lds[(3 * 16 + l15) * LDST + ka + 1] };
      acc0 = wmma4(a, b0, acc0);
      acc1 = wmma4(a, b1, acc1);
      acc2 = wmma4(a, b2, acc2);
      acc3 = wmma4(a, b3, acc3);
    }
#pragma unroll
    for (int v = 0; v < 8; ++v) {
      int y = y0 + v + half * 8;
      float s0 = acc0[v], s1 = acc1[v], s2 = acc2[v], s3 = acc3[v];
      size_t sb = srbase + (size_t)y * 64 + l15;
      s_rel[sb]      = s0;
      s_rel[sb + 16] = s1;
      s_rel[sb + 32] = s2;
      s_rel[sb + 48] = s3;
      // softmax over the 64 r-values of this y-row (spread over 16-lane half)
      float m = fmaxf(fmaxf(s0, s1), fmaxf(s2, s3));
      for (int d = 1; d < 16; d <<= 1) m = fmaxf(m, __shfl_xor(m, d, 32));
      float e0 = __expf(s0 - m), e1 = __expf(s1 - m);
      float e2 = __expf(s2 - m), e3 = __expf(s3 - m);
      float s = e0 + e1 + e2 + e3;
      for (int d = 1; d < 16; d <<= 1) s += __shfl_xor(s, d, 32);
      float w = pa_lds[y] / s;
      accA0 += e0 * w; accA1 += e1 * w; accA2 += e2 * w; accA3 += e3 * w;
    }
  }
  atomicAdd(&as_lds[0 * 16 + l15], accA0);
  atomicAdd(&as_lds[1 * 16 + l15], accA1);
  atomicAdd(&as_lds[2 * 16 + l15], accA2);
  atomicAdd(&as_lds[3 * 16 + l15], accA3);
  __syncthreads();
  if (tid < 64) Asum[(size_t)row * 64 + tid] = as_lds[tid];
}

// ---------------------------------------------------------------------------
// Wsum[k,h] = sum_m head_WV[k,h,m]; one wave per (k,h) row of 512.
// ---------------------------------------------------------------------------
__global__ void wsum_kernel(const float* __restrict__ WV, float* __restrict__ Wsum) {
  int row = blockIdx.x * 8 + (threadIdx.x >> 5);
  int lane = threadIdx.x & 31;
  const float* p = WV + (size_t)row * 512;
  float s = 0.f;
  for (int m = lane; m < 512; m += 32) s += p[m];
  for (int d = 1; d < 32; d <<= 1) s += __shfl_xor(s, d, 32);
  if (lane == 0) Wsum[row] = s;
}

// ---------------------------------------------------------------------------
// out[row,m] = gelu( sum_{col<32768} feat(row,col) * Wd[m,col] + bd[m] )
// feat(row, h*64+k) = x[row,h]*Asum[row,k]*Wsum[k,h], built on the fly from
// LDS-staged x/Asum tiles (feat never touches HBM). 4 waves per 16-row tile.
// ---------------------------------------------------------------------------
__global__ __launch_bounds__(128) void final_gemm(
    const float* __restrict__ x, const float* __restrict__ Asum,
    const float* __restrict__ Wsum, const float* __restrict__ Wd,
    const float* __restrict__ bd, float* __restrict__ out) {
  __shared__ float x_l[16 * 512];
  __shared__ float a_l[16 * 64];
  int m0 = blockIdx.x * 16;
  int tid = threadIdx.x;
  int n0 = blockIdx.y * 64 + (tid >> 5) * 16;
  for (int i = tid; i < 16 * 512; i += 128)
    x_l[i] = x[(size_t)(m0 + (i >> 9)) * 512 + (i & 511)];
  for (int i = tid; i < 16 * 64; i += 128)
    a_l[i] = Asum[(size_t)(m0 + (i >> 6)) * 64 + (i & 63)];
  __syncthreads();

  int lane = tid & 31;
  int half = lane >> 4, l15 = lane & 15;
  const float* wrow = Wd + (size_t)(n0 + l15) * 32768;
  const float* xr = x_l + l15 * 512;
  const float* ar = a_l + l15 * 64;
  v8f acc = vzero8();
  for (int kk = 0; kk < 8192; ++kk) {
    int col = kk * 4 + half * 2;       // col and col+1 share h (R=64, step 4)
    int h = col >> 6;
    int k2 = col & 63;
    float xv = xr[h];
    float a0 = ar[k2], a1 = ar[k2 + 1];
    float w0 = Wsum[k2 * 512 + h], w1 = Wsum[(k2 + 1) * 512 + h];
    v2f a = { xv * a0 * w0, xv * a1 * w1 };
    v2f b = { wrow[col], wrow[col + 1] };
    acc = wmma4(a, b, acc);
  }
#pragma unroll
  for (int v = 0; v < 8; ++v) {
    int m = m0 + v + half * 8;
    int n = n0 + l15;
    float val = acc[v] + bd[n];
    val = 0.5f * val * (1.f + erff(val * 0.70710678118654752f));  // exact gelu
    out[(size_t)m * 512 + n] = val;
  }
}

// ---------------------------------------------------------------------------
extern "C" void kernel_launch(void* const* d_in, const int* in_sizes, int n_in,
                              void* d_out, int out_size, void* d_ws, size_t ws_size,
                              hipStream_t stream) {
  const float* x          = (const float*)d_in[0];
  const int*   mask       = (const int*)d_in[1];
  const float* W_arc_d    = (const float*)d_in[2];
  const float* b_arc_d    = (const float*)d_in[3];
  const float* W_arc_h    = (const float*)d_in[4];
  const float* b_arc_h    = (const float*)d_in[5];
  const float* W_rel_d    = (const float*)d_in[6];
  const float* b_rel_d    = (const float*)d_in[7];
  const float* W_rel_h    = (const float*)d_in[8];
  const float* b_rel_h    = (const float*)d_in[9];
  const float* W_arc_attn = (const float*)d_in[10];
  const float* W_rel_attn = (const float*)d_in[11];
  const float* head_WV    = (const float*)d_in[12];
  const float* W_dense    = (const float*)d_in[13];
  const float* b_dense    = (const float*)d_in[14];

  float* out_sarc = (float*)d_out;                       // 4*512*512
  float* out_srel = out_sarc + (size_t)2048 * 512;       // 4*512*512*64
  float* out_gelu = out_srel + (size_t)2048 * 512 * 64;  // 4*512*512

  float* ws = (float*)d_ws;
  size_t o = 0;
  float* arc_d = ws + o; o += (size_t)2048 * 512;
  float* arc_h = ws + o; o += (size_t)2048 * 512;
  float* Ubuf  = ws + o; o += (size_t)2048 * 512;
  float* Dp    = ws + o; o += (size_t)2048 * 132;   // rel_d1 padded
  float* Hp    = ws + o; o += (size_t)2048 * 132;   // rel_h1 padded
  float* Wp    = ws + o; o += (size_t)64 * 132 * 144;
  float* p_arc = ws + o; o += (size_t)2048 * 512;
  float* Tbuf  = ws + o; o += (size_t)2048 * 64 * 144;
  float* AsumB = ws + o; o += (size_t)2048 * 64;
  float* WsumB = ws + o; o += (size_t)64 * 512;
  (void)o; (void)ws_size; (void)in_sizes; (void)n_in; (void)out_size;

  pad_kernel<<<dim3(4752), dim3(256), 0, stream>>>(W_rel_attn, Wp, Dp, Hp);
  mlp_gemm<<<dim3(128, 32), dim3(32), 0, stream>>>(x, W_arc_d, b_arc_d, arc_d, 512, 512, 1);
  mlp_gemm<<<dim3(128, 32), dim3(32), 0, stream>>>(x, W_arc_h, b_arc_h, arc_h, 512, 512, 1);
  mlp_gemm<<<dim3(128, 8),  dim3(32), 0, stream>>>(x, W_rel_d, b_rel_d, Dp, 512, 132, 1);
  mlp_gemm<<<dim3(128, 8),  dim3(32), 0, stream>>>(x, W_rel_h, b_rel_h, Hp, 512, 132, 1);
  u_gemm<<<dim3(128, 32), dim3(32), 0, stream>>>(arc_d, W_arc_attn, Ubuf);
  sarc_gemm<<<dim3(32, 32, 4), dim3(32), 0, stream>>>(Ubuf, arc_h, mask, out_sarc);
  softmax_arc<<<dim3(2048), dim3(256), 0, stream>>>(out_sarc, mask, p_arc);
  t_gemm<<<dim3(128, 64), dim3(32), 0, stream>>>(Dp, Wp, Tbuf);
  srel_fused<<<dim3(2048), dim3(256), 0, stream>>>(Hp, Tbuf, p_arc, out_srel, AsumB);
  wsum_kernel<<<dim3(4096), dim3(256), 0, stream>>>(head_WV, WsumB);
  final_gemm<<<dim3(128, 8), dim3(128), 0, stream>>>(x, AsumB, WsumB, W_dense, b_dense, out_gelu);
}